// BBLoss_15264313770150
// MI455X (gfx1250) — compile-verified
//
#include <hip/hip_runtime.h>
#include <math.h>

typedef __attribute__((ext_vector_type(2))) float v2f;
typedef __attribute__((ext_vector_type(8))) float v8f;

#define BATCH 4
#define CHAN  3
#define HIN   192
#define N1    4096     // 64*64 full-res patches
#define N2    1024     // 32*32 patches of /2 image
#define N4    256      // 16*16 patches of /4 image
#define MEXT  (N2+N4)  // 1280
#define MTOT  (N1+MEXT)// 5376
#define KP    32       // padded K (27 valid + 5 zeros)
#define KV    27
#define MSPLIT 4                 // waves per block, each scans 1/4 of m-range
#define MT_TILES (MTOT/16)       // 336 m-tiles
#define MT_PER (MT_TILES/MSPLIT) // 84

// workspace layout (float offsets)
#define OFF_P1   0
#define OFF_P2   (OFF_P1 + BATCH*N1*KP)
#define OFF_Y2   (OFF_P2 + BATCH*N1*KP)          // down-sampled patches [B,1280,32]
#define OFF_XN1  (OFF_Y2 + BATCH*MEXT*KP)
#define OFF_XN2  (OFF_XN1 + BATCH*N1)
#define OFF_YM2  (OFF_XN2 + BATCH*N1)
#define OFF_G2   (OFF_YM2 + BATCH*MEXT)
#define OFF_G4   (OFF_G2 + BATCH*CHAN*96*96)
#define OFF_IND  (OFF_G4 + BATCH*CHAN*48*48)
#define OFF_PART (OFF_IND + BATCH*N1)

// ---------------------------------------------------------------------------
// Bicubic downsample by even factor s (a=-0.75, align_corners=False):
// fractional offset is always 0.5 -> fixed 4-tap kernel, separable 4x4.
__global__ void bicubic_down_kernel(const float* __restrict__ src,
                                    float* __restrict__ dst, int Hs, int s) {
  const int Ho = Hs / s;
  const int total = BATCH * CHAN * Ho * Ho;
  int t = blockIdx.x * blockDim.x + threadIdx.x;
  if (t >= total) return;
  int j = t % Ho;
  int i = (t / Ho) % Ho;
  int bc = t / (Ho * Ho);
  const float w[4] = {-0.09375f, 0.59375f, 0.59375f, -0.09375f};
  const int off = (s - 2) >> 1;
  const int bi = s * i + off, bj = s * j + off;
  const float* p = src + (size_t)bc * Hs * Hs;
  float acc = 0.f;
#pragma unroll
  for (int u = 0; u < 4; u++) {
    int r = bi + u - 1; r = r < 0 ? 0 : (r > Hs - 1 ? Hs - 1 : r);
    float ra = 0.f;
#pragma unroll
    for (int v = 0; v < 4; v++) {
      int c = bj + v - 1; c = c < 0 ? 0 : (c > Hs - 1 ? Hs - 1 : c);
      ra += w[v] * p[r * Hs + c];
    }
    acc += w[u] * ra;
  }
  dst[t] = acc;
}

// ---------------------------------------------------------------------------
// Unfold (k=3, stride=3) into padded [rows, 32] matrix + squared-norm vector.
__global__ void unfold_norm_kernel(const float* __restrict__ src, int Hs,
                                   float* __restrict__ P, float* __restrict__ norms,
                                   int rowsPerBatch, int moff) {
  const int side = Hs / 3;
  const int np = side * side;
  int t = blockIdx.x * blockDim.x + threadIdx.x;
  if (t >= BATCH * np) return;
  int b = t / np, n = t % np;
  int ph = n / side, pw = n % side;
  const float* img = src + (size_t)b * CHAN * Hs * Hs;
  float* row = P + (size_t)(b * rowsPerBatch + moff + n) * KP;
  float s2 = 0.f;
  int k = 0;
  for (int c = 0; c < CHAN; c++)
    for (int kh = 0; kh < 3; kh++)
      for (int kw = 0; kw < 3; kw++) {
        float v = img[(size_t)c * Hs * Hs + (3 * ph + kh) * Hs + (3 * pw + kw)];
        row[k++] = v;
        s2 += v * v;
      }
  for (; k < KP; k++) row[k] = 0.f;
  norms[b * rowsPerBatch + moff + n] = s2;
}

// ---------------------------------------------------------------------------
// Fused dual-GEMM + score + argmin. One block per (batch, 16-row n-tile);
// MSPLIT=4 waves each scan an ascending quarter of the 336 m-tiles.
// D-tile orientation: A = p2_cat m-tile (16xK), B = p1/p2 n-tile transposed
// (Kx16) -> D[m,n]: m on VGPR axis (8 per lane, split over lane halves),
// n = lane%16. Argmin over m: per-lane running min -> xor-16 shuffle ->
// LDS combine across the 4 waves in ascending m order (first-min tie-break).
__global__ __launch_bounds__(128) void dist_argmin_kernel(
    const float* __restrict__ ws_f, int* __restrict__ ind) {
  const float* P1  = ws_f + OFF_P1;
  const float* P2  = ws_f + OFF_P2;
  const float* Y2  = ws_f + OFF_Y2;
  const float* XN1 = ws_f + OFF_XN1;
  const float* XN2 = ws_f + OFF_XN2;
  const float* YM2 = ws_f + OFF_YM2;

  __shared__ float shB[MSPLIT * 16];
  __shared__ int   shI[MSPLIT * 16];

  const int b  = blockIdx.x >> 8;    // 256 n-tiles per batch image
  const int nt = blockIdx.x & 255;
  const int n0 = nt * 16;
  const int wv = threadIdx.x >> 5;   // m-quarter index 0..3
  const int lane = threadIdx.x & 31;
  const int lo = lane & 15;
  const int hi = lane >> 4;

  // Preload B operands (transposed 16 n-rows of P1/P2), all 8 K-steps.
  // B layout (4x16, f32): v0 = K{k0+hi}, v1 = K{k0+2+hi}, N = lane%16.
  const float* p1row = P1 + (size_t)(b * N1 + n0 + lo) * KP;
  const float* p2row = P2 + (size_t)(b * N1 + n0 + lo) * KP;
  v2f B1[8], B2[8];
#pragma unroll
  for (int j = 0; j < 8; j++) {
    B1[j].x = p1row[4 * j + hi];     B1[j].y = p1row[4 * j + 2 + hi];
    B2[j].x = p2row[4 * j + hi];     B2[j].y = p2row[4 * j + 2 + hi];
  }
  const float xs1 = XN1[b * N1 + n0 + lo];
  const float xs2 = XN2[b * N1 + n0 + lo];

  float best = 3.4e38f;
  int bestIdx = 0;

  for (int mi = 0; mi < MT_PER; mi++) {
    const int mt = wv * MT_PER + mi;
    const float* Yb; const float* ymb; int r0;
    if (mt < N1 / 16) {        // p2 portion of p2_cat: read P2 directly
      Yb = P2 + (size_t)b * N1 * KP;   ymb = XN2 + b * N1;   r0 = mt * 16;
    } else {                   // downsampled portion
      Yb = Y2 + (size_t)b * MEXT * KP; ymb = YM2 + b * MEXT; r0 = (mt - N1 / 16) * 16;
    }
    // A layout (16x4, f32): lanes 0-15 hold K{4j,4j+1}, lanes 16-31 K{4j+2,4j+3}; M=lane%16.
    const float* yrow = Yb + (size_t)(r0 + lo) * KP;
    v8f acc1 = {}; v8f acc2 = {};
#pragma unroll
    for (int j = 0; j < 8; j++) {
      v2f a = *(const v2f*)(yrow + 4 * j + 2 * hi);
      acc1 = __builtin_amdgcn_wmma_f32_16x16x4_f32(false, a, false, B1[j],
                                                   (short)0, acc1, false, false);
      acc2 = __builtin_amdgcn_wmma_f32_16x16x4_f32(false, a, false, B2[j],
                                                   (short)0, acc2, false, false);
    }
    // D element r <-> m = mt*16 + r + 8*hi ; n = n0 + lo.
    const float* ymp = ymb + r0 + 8 * hi;
#pragma unroll
    for (int r = 0; r < 8; r++) {
      float ymv = ymp[r];
      float d1 = fmaxf(xs1 + ymv - 2.f * acc1[r], 0.f);
      float d2 = fmaxf(xs2 + ymv - 2.f * acc2[r], 0.f);
      float s = d1 + d2;
      int m = mt * 16 + 8 * hi + r;
      if (s < best) { best = s; bestIdx = m; }   // ascending m -> first min kept
    }
  }
  // combine the two lane halves (same n, disjoint m sets); ties -> smaller index
  float ob = __shfl_xor(best, 16, 32);
  int   oi = __shfl_xor(bestIdx, 16, 32);
  if (ob < best || (ob == best && oi < bestIdx)) { best = ob; bestIdx = oi; }
  if (hi == 0) { shB[wv * 16 + lo] = best; shI[wv * 16 + lo] = bestIdx; }
  __syncthreads();
  // combine the 4 m-quarters (ascending order, strict rules -> first min)
  if (threadIdx.x < 16) {
    float bb = shB[threadIdx.x];
    int   bi = shI[threadIdx.x];
#pragma unroll
    for (int w = 1; w < MSPLIT; w++) {
      float owb = shB[w * 16 + threadIdx.x];
      int   owi = shI[w * 16 + threadIdx.x];
      if (owb < bb || (owb == bb && owi < bi)) { bb = owb; bi = owi; }
    }
    ind[b * N1 + n0 + threadIdx.x] = bi;
  }
}

// ---------------------------------------------------------------------------
// Gather selected patch + block-level L1 partial sums (deterministic).
__global__ __launch_bounds__(256) void l1_partial_kernel(
    const float* __restrict__ ws_f, const int* __restrict__ ind,
    float* __restrict__ part) {
  __shared__ float sh[256];
  int t = blockIdx.x * 256 + threadIdx.x;   // over BATCH*N1 = 16384
  int b = t / N1, n = t % N1;
  const float* P1 = ws_f + OFF_P1;
  const float* P2 = ws_f + OFF_P2;
  const float* Y2 = ws_f + OFF_Y2;
  const float* prow = P1 + (size_t)(b * N1 + n) * KP;
  int m = ind[b * N1 + n];
  const float* srow = (m < N1) ? (P2 + (size_t)(b * N1 + m) * KP)
                               : (Y2 + (size_t)(b * MEXT + (m - N1)) * KP);
  float s = 0.f;
#pragma unroll
  for (int k = 0; k < KV; k++) s += fabsf(prow[k] - srow[k]);
  sh[threadIdx.x] = s;
  __syncthreads();
  for (int off = 128; off > 0; off >>= 1) {
    if (threadIdx.x < off) sh[threadIdx.x] += sh[threadIdx.x + off];
    __syncthreads();
  }
  if (threadIdx.x == 0) part[blockIdx.x] = sh[0];
}

__global__ void finalize_kernel(const float* __restrict__ part,
                                float* __restrict__ out) {
  if (threadIdx.x == 0 && blockIdx.x == 0) {
    float s = 0.f;
    for (int i = 0; i < 64; i++) s += part[i];   // fixed order -> deterministic
    out[0] = s / (float)(BATCH * N1 * KV);
  }
}

// ---------------------------------------------------------------------------
extern "C" void kernel_launch(void* const* d_in, const int* in_sizes, int n_in,
                              void* d_out, int out_size, void* d_ws, size_t ws_size,
                              hipStream_t stream) {
  const float* x  = (const float*)d_in[0];
  const float* gt = (const float*)d_in[1];
  float* ws = (float*)d_ws;
  float* g2 = ws + OFF_G2;
  float* g4 = ws + OFF_G4;
  int* ind = (int*)(ws + OFF_IND);
  float* part = ws + OFF_PART;

  // bicubic downsample gt by 2 and 4
  {
    int tot2 = BATCH * CHAN * 96 * 96;
    bicubic_down_kernel<<<(tot2 + 255) / 256, 256, 0, stream>>>(gt, g2, HIN, 2);
    int tot4 = BATCH * CHAN * 48 * 48;
    bicubic_down_kernel<<<(tot4 + 255) / 256, 256, 0, stream>>>(gt, g4, HIN, 4);
  }
  // unfold into padded patch matrices + norms
  unfold_norm_kernel<<<(BATCH * N1 + 255) / 256, 256, 0, stream>>>(
      x, HIN, ws + OFF_P1, ws + OFF_XN1, N1, 0);
  unfold_norm_kernel<<<(BATCH * N1 + 255) / 256, 256, 0, stream>>>(
      gt, HIN, ws + OFF_P2, ws + OFF_XN2, N1, 0);
  unfold_norm_kernel<<<(BATCH * N2 + 255) / 256, 256, 0, stream>>>(
      g2, 96, ws + OFF_Y2, ws + OFF_YM2, MEXT, 0);
  unfold_norm_kernel<<<(BATCH * N4 + 255) / 256, 256, 0, stream>>>(
      g4, 48, ws + OFF_Y2, ws + OFF_YM2, MEXT, N2);
  // fused WMMA distance + argmin: 1024 blocks x 4 waves = 4096 waves
  dist_argmin_kernel<<<BATCH * 256, 128, 0, stream>>>(ws, ind);
  // deterministic mean-L1 of p1 vs selected patch
  l1_partial_kernel<<<64, 256, 0, stream>>>(ws, ind, part);
  finalize_kernel<<<1, 64, 0, stream>>>(part, (float*)d_out);
}